// OverISS_T_33998961115417
// MI455X (gfx1250) — compile-verified
//
#include <hip/hip_runtime.h>
#include <math.h>

// ---------------- problem constants (from reference) ----------------
#define B_      2
#define NCH     6
#define NSRC    2
#define NBAK    4      // NCH - NSRC
#define NF      257
#define NT      1000
#define NTAPS   5
#define NPAD    6      // N_TAPS + N_DELAY
#define NROWS   36     // NCH + NCH*NTAPS  (X rows + X_bar rows)
#define NHT     30     // NCH*NTAPS
#define EPSV        1e-3f
#define MODEL_EPSV  1e-5f
#define PB_EPSV     1e-6f

typedef float v8f __attribute__((ext_vector_type(8)));
typedef float v2f __attribute__((ext_vector_type(2)));

#if defined(__HIP_DEVICE_COMPILE__) && __has_builtin(__builtin_amdgcn_wmma_f32_16x16x4_f32)
#define HAVE_WMMA_F32 1
#else
#define HAVE_WMMA_F32 0
#endif

// ---------------- complex helpers (float2) ----------------
__device__ __forceinline__ float2 cmul(float2 a, float2 b) {
  return make_float2(a.x * b.x - a.y * b.y, a.x * b.y + a.y * b.x);
}
__device__ __forceinline__ float2 cadd(float2 a, float2 b) { return make_float2(a.x + b.x, a.y + b.y); }
__device__ __forceinline__ float2 csub(float2 a, float2 b) { return make_float2(a.x - b.x, a.y - b.y); }
__device__ __forceinline__ float2 cdiv(float2 a, float2 b) {
  float inv = 1.0f / (b.x * b.x + b.y * b.y);
  return make_float2((a.x * b.x + a.y * b.y) * inv, (a.y * b.x - a.x * b.y) * inv);
}

// =====================================================================
// K0: per-(b,f): Gram matrices C_XX (6x6) and C_XbarX (30x6) via f32 WMMA,
//     plus init of Y = X[:NSRC], W = I rows, H = 0.
// Rows R[r][n]: r<6 -> X[r][n]; r>=6 -> X[c][n-6+t] (c=(r-6)/5, t=(r-6)%5).
// C[r][d] = (1/NT) * sum_n R[r][n] * conj(X[d][n]).
// Out-of-range rows/cols read from a shared zero row -> no predicated loads.
// =====================================================================
__global__ __launch_bounds__(256) void k_init_corr(
    const float* __restrict__ Xr, const float* __restrict__ Xi,
    float2* __restrict__ Yg, float2* __restrict__ Wg, float2* __restrict__ Hg,
    float2* __restrict__ CXX, float2* __restrict__ CXB)
{
  __shared__ __align__(16) float xr[NCH][NT + 8];
  __shared__ __align__(16) float xi[NCH][NT + 8];
  __shared__ __align__(16) float zrow[NT + 8];     // all zeros
#if HAVE_WMMA_F32
  __shared__ float cre[3][16][16];
  __shared__ float cim[3][16][16];
#endif
  const int tid = threadIdx.x;
  const int bf = blockIdx.x;
  const int b = bf / NF, f = bf % NF;

  // load X with NPAD leading zeros (this realizes X_pad / X_bar indexing)
  for (int idx = tid; idx < NCH * (NT + 8); idx += 256) {
    int c = idx / (NT + 8), n = idx % (NT + 8);
    float vr = 0.f, vi = 0.f;
    if (n >= NPAD && n < NPAD + NT) {
      size_t g = ((size_t)(b * NCH + c) * NF + f) * NT + (n - NPAD);
      vr = Xr[g]; vi = Xi[g];
    }
    xr[c][n] = vr; xi[c][n] = vi;
  }
  for (int n = tid; n < NT + 8; n += 256) zrow[n] = 0.f;
  __syncthreads();

#if HAVE_WMMA_F32
  {
    const int wave = tid >> 5, lane = tid & 31;
    const int tile = wave >> 1, half = wave & 1;   // 3 M-tiles x 2 K-halves; waves 6,7 idle
    const bool active = (tile < 3);
    v8f aRe = {0, 0, 0, 0, 0, 0, 0, 0};
    v8f aP  = {0, 0, 0, 0, 0, 0, 0, 0};
    v8f aM  = {0, 0, 0, 0, 0, 0, 0, 0};
    if (active) {
      // A fragment (32-bit A 16x4): lane m = row, lane-group picks K pair
      const int m  = lane & 15;
      const int kb = (lane >> 4) << 1;
      const int r  = tile * 16 + m;
      const float* par = &zrow[0];
      const float* pai = &zrow[0];
      if (r < NCH)        { par = &xr[r][NPAD]; pai = &xi[r][NPAD]; }
      else if (r < NROWS) {
        int c = (r - NCH) / NTAPS, t = (r - NCH) % NTAPS;
        par = &xr[c][t]; pai = &xi[c][t];
      }
      const int d = m;                // B fragment: lane group -> K pair, lane -> column
      const float* pbr = (d < NCH) ? &xr[d][NPAD] : &zrow[0];
      const float* pbi = (d < NCH) ? &xi[d][NPAD] : &zrow[0];
      for (int n0 = half * 500; n0 < half * 500 + 500; n0 += 4) {
        const int n = n0 + kb;                     // n even; NPAD even => B 8B-aligned
        v2f ar, ai;
        ar[0] = par[n]; ar[1] = par[n + 1];
        ai[0] = pai[n]; ai[1] = pai[n + 1];
        v2f br = *(const v2f*)(pbr + n);
        v2f bi = *(const v2f*)(pbi + n);
        // Re += Rr*Xr + Ri*Xi ; Im = (Ri*Xr) - (Rr*Xi), subtraction at writeout
        aRe = __builtin_amdgcn_wmma_f32_16x16x4_f32(false, ar, false, br, (short)0, aRe, false, false);
        aRe = __builtin_amdgcn_wmma_f32_16x16x4_f32(false, ai, false, bi, (short)0, aRe, false, false);
        aP  = __builtin_amdgcn_wmma_f32_16x16x4_f32(false, ai, false, br, (short)0, aP,  false, false);
        aM  = __builtin_amdgcn_wmma_f32_16x16x4_f32(false, ar, false, bi, (short)0, aM,  false, false);
      }
    }
    // combine the two K-halves of each tile through LDS
    if (active && half == 0) {
#pragma unroll
      for (int j = 0; j < 8; ++j) {
        int M = ((lane >> 4) << 3) + j, N = lane & 15;  // C/D layout: VGPR j -> M = j + 8*(lane>=16)
        cre[tile][M][N] = aRe[j];
        cim[tile][M][N] = aP[j] - aM[j];
      }
    }
    __syncthreads();
    if (active && half == 1) {
#pragma unroll
      for (int j = 0; j < 8; ++j) {
        int M = ((lane >> 4) << 3) + j, N = lane & 15;
        cre[tile][M][N] += aRe[j];
        cim[tile][M][N] += aP[j] - aM[j];
      }
    }
    __syncthreads();
  }
#endif

  // write C_XX / C_XbarX
  for (int idx = tid; idx < NROWS * NCH; idx += 256) {
    int r = idx / NCH, d = idx % NCH;
    float sre, sim;
#if HAVE_WMMA_F32
    sre = cre[r >> 4][r & 15][d] * (1.0f / NT);
    sim = cim[r >> 4][r & 15][d] * (1.0f / NT);
#else
    const float* par; const float* pai;
    if (r < NCH) { par = &xr[r][NPAD]; pai = &xi[r][NPAD]; }
    else {
      int c = (r - NCH) / NTAPS, t = (r - NCH) % NTAPS;
      par = &xr[c][t]; pai = &xi[c][t];
    }
    const float* pbr = &xr[d][NPAD]; const float* pbi = &xi[d][NPAD];
    float are = 0.f, aim = 0.f;
    for (int n = 0; n < NT; ++n) {
      float rr = par[n], ri = pai[n], brv = pbr[n], biv = pbi[n];
      are += rr * brv + ri * biv;
      aim += ri * brv - rr * biv;
    }
    sre = are * (1.0f / NT); sim = aim * (1.0f / NT);
#endif
    float2 val = make_float2(sre, sim);
    if (r < NCH) CXX[((size_t)(b * NF + f) * NCH + r) * NCH + d] = val;
    else         CXB[((size_t)(b * NF + f) * NHT + (r - NCH)) * NCH + d] = val;
  }
  // init Y = X[:NSRC]
  for (int idx = tid; idx < NSRC * NT; idx += 256) {
    int s = idx / NT, n = idx % NT;
    Yg[((size_t)(b * NSRC + s) * NF + f) * NT + n] = make_float2(xr[s][n + NPAD], xi[s][n + NPAD]);
  }
  // init W = eye rows, H = 0
  for (int idx = tid; idx < NSRC * NCH; idx += 256) {
    int s = idx / NCH, d = idx % NCH;
    Wg[((size_t)(b * NSRC + s) * NF + f) * NCH + d] = make_float2((s == d) ? 1.f : 0.f, 0.f);
  }
  for (int idx = tid; idx < NSRC * NHT; idx += 256) {
    int s = idx / NHT, r2 = idx % NHT;
    Hg[((size_t)(b * NSRC + s) * NF + f) * NHT + r2] = make_float2(0.f, 0.f);
  }
}

// =====================================================================
// K_zero / K1: cross-frequency reductions for Laplace weights:
//   S[b,s,n] = sum_f |Y|^2 ;  G[b,s] = sum_{f,n} |Y|^2
// =====================================================================
__global__ void k_zero_g(float* G) {
  if (threadIdx.x < B_ * NSRC) G[threadIdx.x] = 0.f;
}

__global__ __launch_bounds__(256) void k_wsum(const float2* __restrict__ Yg,
                                              float* __restrict__ S,
                                              float* __restrict__ G)
{
  __shared__ float sr[256];
  const int bs = blockIdx.x >> 2;                 // (b*NSRC + s)
  const int n = ((blockIdx.x & 3) << 8) + threadIdx.x;
  float acc = 0.f;
  if (n < NT) {
    size_t base = (size_t)bs * NF * NT + n;
    for (int f = 0; f < NF; ++f) {
      float2 y = Yg[base + (size_t)f * NT];
      acc += y.x * y.x + y.y * y.y;
    }
    S[bs * NT + n] = acc;
  }
  sr[threadIdx.x] = acc;
  __syncthreads();
  for (int s2 = 128; s2 > 0; s2 >>= 1) {
    if (threadIdx.x < s2) sr[threadIdx.x] += sr[threadIdx.x + s2];
    __syncthreads();
  }
  if (threadIdx.x == 0) atomicAdd(&G[bs], sr[0]);
}

// =====================================================================
// K2: one full ISS iteration per (b,f). Everything resident in LDS.
// Dynamic LDS layout: xs[6][1008] f2 | ys[2][1000] f2 | zs[4][1000] f2 |
//                     wgt[2][1000] f | red[256][6] f      (= 110528 B)
// =====================================================================
#define K2_SMEM ((NCH*(NT+8) + NSRC*NT + NBAK*NT) * (int)sizeof(float2) + \
                 (NSRC*NT + 256*6) * (int)sizeof(float))

__global__ __launch_bounds__(256) void k_iter(
    const float* __restrict__ Xr, const float* __restrict__ Xi,
    float2* __restrict__ Yg, float2* __restrict__ Wg, float2* __restrict__ Hg,
    const float2* __restrict__ CXX, const float2* __restrict__ CXB,
    const float* __restrict__ S, const float* __restrict__ G)
{
  extern __shared__ char smem_raw[];
  float2* xs  = (float2*)smem_raw;          // [NCH][NT+8], zero-padded front
  float2* ysv = xs + NCH * (NT + 8);        // [NSRC][NT]
  float2* zsv = ysv + NSRC * NT;            // [NBAK][NT]
  float*  wgt = (float*)(zsv + NBAK * NT);  // [NSRC][NT]
  float*  red = wgt + NSRC * NT;            // [256][6]

  __shared__ float2 Wl[NSRC][NCH];
  __shared__ float2 Hl[NSRC][NCH][NTAPS];
  __shared__ float2 Jl[NBAK][NSRC];
  __shared__ float2 Am[NSRC][NCH];
  __shared__ float2 vsh[NSRC];
  __shared__ float2 tmpW[NCH];
  __shared__ float2 tmpH[NCH][NTAPS];
  __shared__ float  ginv[NSRC];
  __shared__ float  gval[NSRC];

  const int tid = threadIdx.x;
  const int bf = blockIdx.x;
  const int b = bf / NF, f = bf % NF;

  if (tid < NSRC) {
    float g = fmaxf(G[b * NSRC + tid] / (float)(NF * NT), EPSV);
    gval[tid] = g;
    ginv[tid] = 1.0f / fmaxf(sqrtf(g), EPSV);
  }
  __syncthreads();

  // ---- loads (fused with g-rescale of Y, W, H; weights = g / max(2*sqrt(S), 1e-5)) ----
  for (int idx = tid; idx < NCH * (NT + 8); idx += 256) {
    int c = idx / (NT + 8), n = idx % (NT + 8);
    float2 v = make_float2(0.f, 0.f);
    if (n >= NPAD && n < NPAD + NT) {
      size_t g2 = ((size_t)(b * NCH + c) * NF + f) * NT + (n - NPAD);
      v = make_float2(Xr[g2], Xi[g2]);
    }
    xs[c * (NT + 8) + n] = v;
  }
  for (int idx = tid; idx < NSRC * NT; idx += 256) {
    int s = idx / NT, n = idx % NT;
    float2 y = Yg[((size_t)(b * NSRC + s) * NF + f) * NT + n];
    ysv[s * NT + n] = make_float2(y.x * ginv[s], y.y * ginv[s]);
    wgt[s * NT + n] = gval[s] / fmaxf(2.0f * sqrtf(S[(b * NSRC + s) * NT + n]), MODEL_EPSV);
  }
  if (tid < NSRC * NCH) {
    int s = tid / NCH, d = tid % NCH;
    float2 w = Wg[((size_t)(b * NSRC + s) * NF + f) * NCH + d];
    Wl[s][d] = make_float2(w.x * ginv[s], w.y * ginv[s]);
  }
  if (tid >= 64 && tid < 64 + NSRC * NHT) {
    int i2 = tid - 64;
    int s = i2 / NHT, r2 = i2 % NHT;
    float2 h = Hg[((size_t)(b * NSRC + s) * NF + f) * NHT + r2];
    Hl[s][r2 / NTAPS][r2 % NTAPS] = make_float2(h.x * ginv[s], h.y * ginv[s]);
  }
  __syncthreads();

  // ---- background update: A = W*C_XX + H*C_XbarX ; J = conj(solve(A[:,:2], A[:,2:])) ----
  if (tid < NSRC * NCH) {
    int s = tid / NCH, d = tid % NCH;
    const float2* cxx = &CXX[(size_t)(b * NF + f) * NCH * NCH];
    const float2* cxb = &CXB[(size_t)(b * NF + f) * NHT * NCH];
    float2 acc = make_float2(0.f, 0.f);
    for (int c = 0; c < NCH; ++c) acc = cadd(acc, cmul(Wl[s][c], cxx[c * NCH + d]));
    for (int r2 = 0; r2 < NHT; ++r2) acc = cadd(acc, cmul(Hl[s][r2 / NTAPS][r2 % NTAPS], cxb[r2 * NCH + d]));
    Am[s][d] = acc;
  }
  __syncthreads();
  if (tid == 0) {
    float2 a00 = Am[0][0], a01 = Am[0][1], a10 = Am[1][0], a11 = Am[1][1];
    float2 det = csub(cmul(a00, a11), cmul(a01, a10));
    for (int k = 0; k < NBAK; ++k) {
      float2 r0 = Am[0][NSRC + k], r1 = Am[1][NSRC + k];
      float2 jh0 = cdiv(csub(cmul(a11, r0), cmul(a01, r1)), det);
      float2 jh1 = cdiv(csub(cmul(a00, r1), cmul(a10, r0)), det);
      Jl[k][0] = make_float2(jh0.x, -jh0.y);
      Jl[k][1] = make_float2(jh1.x, -jh1.y);
    }
  }
  __syncthreads();

  // ---- Z = J @ X[:2] - X[2:] ----
  for (int idx = tid; idx < NBAK * NT; idx += 256) {
    int k = idx / NT, n = idx % NT;
    float2 z = cadd(cmul(Jl[k][0], xs[0 * (NT + 8) + n + NPAD]),
                    cmul(Jl[k][1], xs[1 * (NT + 8) + n + NPAD]));
    zsv[k * NT + n] = csub(z, xs[(NSRC + k) * (NT + 8) + n + NPAD]);
  }
  __syncthreads();

  // per-sweep statistics: [vnum0.re, vnum0.im, vnum1.re, vnum1.im, vden0, vden1]
  auto reduce6 = [&](const float2* Zv) {
    float p0 = 0, p1 = 0, p2 = 0, p3 = 0, p4 = 0, p5 = 0;
    for (int n = tid; n < NT; n += 256) {
      float2 z = Zv[n];
      float zz = z.x * z.x + z.y * z.y;
      float w0 = wgt[n], w1 = wgt[NT + n];
      float2 y0 = ysv[n], y1 = ysv[NT + n];
      p0 += w0 * (y0.x * z.x + y0.y * z.y);
      p1 += w0 * (y0.y * z.x - y0.x * z.y);
      p2 += w1 * (y1.x * z.x + y1.y * z.y);
      p3 += w1 * (y1.y * z.x - y1.x * z.y);
      p4 += w0 * zz;
      p5 += w1 * zz;
    }
    red[tid * 6 + 0] = p0; red[tid * 6 + 1] = p1; red[tid * 6 + 2] = p2;
    red[tid * 6 + 3] = p3; red[tid * 6 + 4] = p4; red[tid * 6 + 5] = p5;
    __syncthreads();
    for (int s2 = 128; s2 > 0; s2 >>= 1) {
      if (tid < s2) {
#pragma unroll
        for (int j = 0; j < 6; ++j) red[tid * 6 + j] += red[(tid + s2) * 6 + j];
      }
      __syncthreads();
    }
  };
  auto apply_y = [&](const float2* Zv) {
    for (int n = tid; n < NT; n += 256) {
      float2 z = Zv[n];
      ysv[n]      = csub(ysv[n],      cmul(vsh[0], z));
      ysv[NT + n] = csub(ysv[NT + n], cmul(vsh[1], z));
    }
  };

  // ---- source sweeps (iss_v1) ----
  for (int src = 0; src < NSRC; ++src) {
    reduce6(&ysv[src * NT]);
    if (tid == 0) {
      const float invN = 1.0f / (float)NT;
      float2 vv[2];
      for (int c = 0; c < 2; ++c) {
        float den = fmaxf(red[4 + c] * invN, EPSV);
        vv[c] = make_float2(red[2 * c] * invN / den, red[2 * c + 1] * invN / den);
      }
      float dsrc = fmaxf(red[4 + src] * invN, EPSV);
      vv[src] = make_float2(1.0f - 1.0f / sqrtf(dsrc), 0.f);
      vsh[0] = vv[0]; vsh[1] = vv[1];
    }
    if (tid >= 32 && tid < 32 + NCH) tmpW[tid - 32] = Wl[src][tid - 32];
    if (tid >= 64 && tid < 64 + NHT) {
      int i2 = tid - 64;
      tmpH[i2 / NTAPS][i2 % NTAPS] = Hl[src][i2 / NTAPS][i2 % NTAPS];
    }
    __syncthreads();
    apply_y(&ysv[src * NT]);   // each n: read old, then update both channels
    if (tid < NSRC * NCH) {
      int c = tid / NCH, d = tid % NCH;
      Wl[c][d] = csub(Wl[c][d], cmul(vsh[c], tmpW[d]));
    }
    if (tid >= 64 && tid < 64 + NSRC * NHT) {
      int i2 = tid - 64; int c = i2 / NHT, r2 = i2 % NHT;
      Hl[c][r2 / NTAPS][r2 % NTAPS] =
          csub(Hl[c][r2 / NTAPS][r2 % NTAPS], cmul(vsh[c], tmpH[r2 / NTAPS][r2 % NTAPS]));
    }
    __syncthreads();
  }

  // ---- background sweeps (iss_v23 with Z) ----
  for (int k = 0; k < NBAK; ++k) {
    reduce6(&zsv[k * NT]);
    if (tid == 0) {
      for (int c = 0; c < 2; ++c) {
        float den = fmaxf(red[4 + c], EPSV);
        vsh[c] = make_float2(red[2 * c] / den, red[2 * c + 1] / den);
      }
    }
    __syncthreads();
    apply_y(&zsv[k * NT]);
    if (tid < NSRC * NSRC) {
      int c = tid / NSRC, d = tid % NSRC;
      Wl[c][d] = csub(Wl[c][d], cmul(vsh[c], Jl[k][d]));
    }
    if (tid >= 32 && tid < 32 + NSRC) {
      int c = tid - 32;
      Wl[c][NSRC + k] = cadd(Wl[c][NSRC + k], vsh[c]);
    }
    __syncthreads();
  }

  // ---- tap sweeps (iss_v23 with X_bar[sc, tap]) ----
  for (int sc = 0; sc < NCH; ++sc) {
    for (int t = 0; t < NTAPS; ++t) {
      const float2* Zv = &xs[sc * (NT + 8) + t];   // Zv[n] = X[sc][n-6+t] via zero pad
      reduce6(Zv);
      if (tid == 0) {
        for (int c = 0; c < 2; ++c) {
          float den = fmaxf(red[4 + c], EPSV);
          vsh[c] = make_float2(red[2 * c] / den, red[2 * c + 1] / den);
        }
      }
      __syncthreads();
      apply_y(Zv);
      if (tid < NSRC) Hl[tid][sc][t] = cadd(Hl[tid][sc][t], vsh[tid]);
      __syncthreads();
    }
  }

  // ---- write back ----
  for (int idx = tid; idx < NSRC * NT; idx += 256) {
    int s = idx / NT, n = idx % NT;
    Yg[((size_t)(b * NSRC + s) * NF + f) * NT + n] = ysv[s * NT + n];
  }
  if (tid < NSRC * NCH) {
    int s = tid / NCH, d = tid % NCH;
    Wg[((size_t)(b * NSRC + s) * NF + f) * NCH + d] = Wl[s][d];
  }
  if (tid >= 64 && tid < 64 + NSRC * NHT) {
    int i2 = tid - 64; int s = i2 / NHT, r2 = i2 % NHT;
    Hg[((size_t)(b * NSRC + s) * NF + f) * NHT + r2] = Hl[s][r2 / NTAPS][r2 % NTAPS];
  }
}

// =====================================================================
// K3: final J, Y = demix_derev, projection back, write stacked re/im.
// =====================================================================
__global__ __launch_bounds__(256) void k_final(
    const float* __restrict__ Xr, const float* __restrict__ Xi,
    const float2* __restrict__ Wg, const float2* __restrict__ Hg,
    const float2* __restrict__ CXX, const float2* __restrict__ CXB,
    float* __restrict__ out)
{
  __shared__ float2 xs[NCH][NT + 8];
  __shared__ float2 Wl[NSRC][NCH];
  __shared__ float2 Hl[NSRC][NCH][NTAPS];
  __shared__ float2 Jl[NBAK][NSRC];
  __shared__ float2 Am[NSRC][NCH];
  __shared__ float2 asc[NSRC];
  const int tid = threadIdx.x;
  const int bf = blockIdx.x;
  const int b = bf / NF, f = bf % NF;

  for (int idx = tid; idx < NCH * (NT + 8); idx += 256) {
    int c = idx / (NT + 8), n = idx % (NT + 8);
    float2 v = make_float2(0.f, 0.f);
    if (n >= NPAD && n < NPAD + NT) {
      size_t g2 = ((size_t)(b * NCH + c) * NF + f) * NT + (n - NPAD);
      v = make_float2(Xr[g2], Xi[g2]);
    }
    xs[c][n] = v;
  }
  if (tid < NSRC * NCH) {
    int s = tid / NCH, d = tid % NCH;
    Wl[s][d] = Wg[((size_t)(b * NSRC + s) * NF + f) * NCH + d];
  }
  if (tid >= 64 && tid < 64 + NSRC * NHT) {
    int i2 = tid - 64; int s = i2 / NHT, r2 = i2 % NHT;
    Hl[s][r2 / NTAPS][r2 % NTAPS] = Hg[((size_t)(b * NSRC + s) * NF + f) * NHT + r2];
  }
  __syncthreads();
  if (tid < NSRC * NCH) {
    int s = tid / NCH, d = tid % NCH;
    const float2* cxx = &CXX[(size_t)(b * NF + f) * NCH * NCH];
    const float2* cxb = &CXB[(size_t)(b * NF + f) * NHT * NCH];
    float2 acc = make_float2(0.f, 0.f);
    for (int c = 0; c < NCH; ++c) acc = cadd(acc, cmul(Wl[s][c], cxx[c * NCH + d]));
    for (int r2 = 0; r2 < NHT; ++r2) acc = cadd(acc, cmul(Hl[s][r2 / NTAPS][r2 % NTAPS], cxb[r2 * NCH + d]));
    Am[s][d] = acc;
  }
  __syncthreads();
  if (tid == 0) {
    float2 a00 = Am[0][0], a01 = Am[0][1], a10 = Am[1][0], a11 = Am[1][1];
    float2 det = csub(cmul(a00, a11), cmul(a01, a10));
    for (int k = 0; k < NBAK; ++k) {
      float2 r0 = Am[0][NSRC + k], r1 = Am[1][NSRC + k];
      float2 jh0 = cdiv(csub(cmul(a11, r0), cmul(a01, r1)), det);
      float2 jh1 = cdiv(csub(cmul(a00, r1), cmul(a10, r0)), det);
      Jl[k][0] = make_float2(jh0.x, -jh0.y);
      Jl[k][1] = make_float2(jh1.x, -jh1.y);
    }
    // projection back: B[s][k] = W1[s][k] + sum_c W2[s][c] * J[c][k]
    float2 Bm[2][2];
    for (int s = 0; s < 2; ++s)
      for (int k = 0; k < 2; ++k) {
        float2 acc = Wl[s][k];
        for (int c = 0; c < NBAK; ++c) acc = cadd(acc, cmul(Wl[s][NSRC + c], Jl[c][k]));
        Bm[s][k] = acc;
      }
    // M[c][s] = B[s][c] + eps*I ; a = M^{-1} e0
    float2 M00 = cadd(Bm[0][0], make_float2(PB_EPSV, 0.f));
    float2 M01 = Bm[1][0];
    float2 M10 = Bm[0][1];
    float2 M11 = cadd(Bm[1][1], make_float2(PB_EPSV, 0.f));
    float2 pdet = csub(cmul(M00, M11), cmul(M01, M10));
    asc[0] = cdiv(M11, pdet);
    asc[1] = cdiv(make_float2(-M10.x, -M10.y), pdet);
  }
  __syncthreads();

  for (int n = tid; n < NT; n += 256) {
    for (int s = 0; s < NSRC; ++s) {
      float2 acc = make_float2(0.f, 0.f);
      for (int d = 0; d < NCH; ++d) acc = cadd(acc, cmul(Wl[s][d], xs[d][n + NPAD]));
      for (int d = 0; d < NCH; ++d)
        for (int t = 0; t < NTAPS; ++t)
          acc = csub(acc, cmul(Hl[s][d][t], xs[d][n + t]));
      float2 yv = cmul(acc, asc[s]);
      size_t o = (((size_t)(b * NSRC + s) * NF + f) * NT + n) * 2;
      out[o] = yv.x; out[o + 1] = yv.y;
    }
  }
}

// =====================================================================
// Host orchestration (all on `stream`, graph-capture safe)
// =====================================================================
extern "C" void kernel_launch(void* const* d_in, const int* in_sizes, int n_in,
                              void* d_out, int out_size, void* d_ws, size_t ws_size,
                              hipStream_t stream) {
  (void)in_sizes; (void)n_in; (void)out_size; (void)ws_size;
  const float* Xr = (const float*)d_in[0];
  const float* Xi = (const float*)d_in[1];
  float* out = (float*)d_out;

  char* ws = (char*)d_ws;
  size_t off = 0;
  auto alloc = [&](size_t bytes) -> void* {
    off = (off + 255) & ~(size_t)255;
    void* p = ws + off;
    off += bytes;
    return p;
  };
  float2* Yg  = (float2*)alloc((size_t)B_ * NSRC * NF * NT * sizeof(float2));
  float2* Wg  = (float2*)alloc((size_t)B_ * NSRC * NF * NCH * sizeof(float2));
  float2* Hg  = (float2*)alloc((size_t)B_ * NSRC * NF * NHT * sizeof(float2));
  float2* CXX = (float2*)alloc((size_t)B_ * NF * NCH * NCH * sizeof(float2));
  float2* CXB = (float2*)alloc((size_t)B_ * NF * NHT * NCH * sizeof(float2));
  float*  S   = (float*)alloc((size_t)B_ * NSRC * NT * sizeof(float));
  float*  G   = (float*)alloc((size_t)B_ * NSRC * sizeof(float));

  // opt in to >64KB dynamic LDS for the ISS iteration kernel (not a stream op;
  // safe under graph capture, idempotent and deterministic)
  (void)hipFuncSetAttribute((const void*)k_iter,
                            hipFuncAttributeMaxDynamicSharedMemorySize, K2_SMEM);

  dim3 blk(256);
  dim3 grd(B_ * NF);

  k_init_corr<<<grd, blk, 0, stream>>>(Xr, Xi, Yg, Wg, Hg, CXX, CXB);
  for (int it = 0; it < 2; ++it) {   // N_ITER = 2
    k_zero_g<<<1, 64, 0, stream>>>(G);
    k_wsum<<<B_ * NSRC * 4, blk, 0, stream>>>(Yg, S, G);
    k_iter<<<grd, blk, K2_SMEM, stream>>>(Xr, Xi, Yg, Wg, Hg, CXX, CXB, S, G);
  }
  k_final<<<grd, blk, 0, stream>>>(Xr, Xi, Wg, Hg, CXX, CXB, out);
}